// ODEBlock_34900904247575
// MI455X (gfx1250) — compile-verified
//
#include <hip/hip_runtime.h>

typedef float v2f __attribute__((ext_vector_type(2)));
typedef float v8f __attribute__((ext_vector_type(8)));

#define BM 64      // batch rows per workgroup
#define DD 64      // state dim
#define HH 128     // hidden dim
#define DP 68      // padded LDS row stride for D-wide buffers (68 mod 64 == 4 -> conflict-free)
#define HP 132     // padded LDS row stride for H-wide buffer  (132 mod 64 == 4)
#define STEPS_PER_SEG 4
#define NSEG 5

// dt is constant: (0.2 - 0.0)/4 = 0.05 for every step of every segment
constexpr double DTd = 0.05;
constexpr float C21 = (float)(DTd * (1.0/5.0));
constexpr float C31 = (float)(DTd * (3.0/40.0));
constexpr float C32 = (float)(DTd * (9.0/40.0));
constexpr float C41 = (float)(DTd * (44.0/45.0));
constexpr float C42 = (float)(DTd * (-56.0/15.0));
constexpr float C43 = (float)(DTd * (32.0/9.0));
constexpr float C51 = (float)(DTd * (19372.0/6561.0));
constexpr float C52 = (float)(DTd * (-25360.0/2187.0));
constexpr float C53 = (float)(DTd * (64448.0/6561.0));
constexpr float C54 = (float)(DTd * (-212.0/729.0));
constexpr float C61 = (float)(DTd * (9017.0/3168.0));
constexpr float C62 = (float)(DTd * (-355.0/33.0));
constexpr float C63 = (float)(DTd * (46732.0/5247.0));
constexpr float C64 = (float)(DTd * (49.0/176.0));
constexpr float C65 = (float)(DTd * (-5103.0/18656.0));
constexpr float DB1 = (float)(DTd * (35.0/384.0));
constexpr float DB3 = (float)(DTd * (500.0/1113.0));
constexpr float DB4 = (float)(DTd * (125.0/192.0));
constexpr float DB5 = (float)(DTd * (-2187.0/6784.0));
constexpr float DB6 = (float)(DTd * (11.0/84.0));

__device__ __forceinline__ float fast_tanh(float x) {
    // tanh(x) = 1 - 2/(exp(2x)+1); exact saturation at +-1, hw-exp accuracy
    float e = __expf(2.0f * x);
    return 1.0f - 2.0f * __builtin_amdgcn_rcpf(e + 1.0f);
}

// One evaluation of f(src) -> dst, using register-resident weight B-fragments.
// FUSE: dst[i] = addsrc[i] + addscale * f(src)[i]  (folds k6 into the y update)
template<bool FUSE>
__device__ __forceinline__ void feval(
    const float* __restrict__ src,   // LDS [BM][DP]
    float*       __restrict__ dst,   // LDS [BM][DP]
    float*       __restrict__ hbuf,  // LDS [BM][HP]
    const v2f (&w1f)[16], const v2f (&w2f)[32],
    float bias1, float bias2,
    int lo, int hi, int n0, int n0b, int mtb,
    const float* __restrict__ addsrc, float addscale)
{
    __syncthreads();  // src fully written

    // ---- GEMM1: Z = src(64x64) @ W1(64x128), tanh, -> hbuf ----
    #pragma unroll
    for (int mt = 0; mt < 4; ++mt) {
        v8f acc0 = {bias1, bias1, bias1, bias1, bias1, bias1, bias1, bias1};
        v8f acc1 = {0.f, 0.f, 0.f, 0.f, 0.f, 0.f, 0.f, 0.f};
        const float* arow = src + (mt * 16 + lo) * DP + 2 * hi;
        #pragma unroll
        for (int kb = 0; kb < 16; kb += 2) {
            v2f a0 = *(const v2f*)(arow + kb * 4);
            v2f a1 = *(const v2f*)(arow + kb * 4 + 4);
            acc0 = __builtin_amdgcn_wmma_f32_16x16x4_f32(
                false, a0, false, w1f[kb], (short)0, acc0, false, false);
            acc1 = __builtin_amdgcn_wmma_f32_16x16x4_f32(
                false, a1, false, w1f[kb + 1], (short)0, acc1, false, false);
        }
        float* hrow = hbuf + (mt * 16 + 8 * hi) * HP + n0 + lo;
        #pragma unroll
        for (int v = 0; v < 8; ++v)
            hrow[v * HP] = fast_tanh(acc0[v] + acc1[v]);
    }

    __syncthreads();  // hbuf fully written

    // ---- GEMM2: F = hbuf(64x128) @ W2(128x64) -> dst ----
    #pragma unroll
    for (int t2 = 0; t2 < 2; ++t2) {
        int mt = mtb + t2;
        v8f acc0 = {bias2, bias2, bias2, bias2, bias2, bias2, bias2, bias2};
        v8f acc1 = {0.f, 0.f, 0.f, 0.f, 0.f, 0.f, 0.f, 0.f};
        const float* arow = hbuf + (mt * 16 + lo) * HP + 2 * hi;
        #pragma unroll
        for (int kb = 0; kb < 32; kb += 2) {
            v2f a0 = *(const v2f*)(arow + kb * 4);
            v2f a1 = *(const v2f*)(arow + kb * 4 + 4);
            acc0 = __builtin_amdgcn_wmma_f32_16x16x4_f32(
                false, a0, false, w2f[kb], (short)0, acc0, false, false);
            acc1 = __builtin_amdgcn_wmma_f32_16x16x4_f32(
                false, a1, false, w2f[kb + 1], (short)0, acc1, false, false);
        }
        int base = (mt * 16 + 8 * hi) * DP + n0b + lo;
        #pragma unroll
        for (int v = 0; v < 8; ++v) {
            float r = acc0[v] + acc1[v];
            if (FUSE)
                dst[base + v * DP] = addsrc[base + v * DP] + addscale * r;
            else
                dst[base + v * DP] = r;
        }
    }

    __syncthreads();  // dst fully written
}

// dst[i] = y[i] + sum_j cs[j]*ks[j][i] over the 64x64 tile, float2-vectorized.
// Identical per-thread element ownership across all passes -> no extra barriers
// between elementwise passes touching the same buffers.
template<int N>
__device__ __forceinline__ void combine(
    float* __restrict__ dst, const float* __restrict__ ybuf,
    const float* const* ks, const float* cs, int tid)
{
    #pragma unroll 2
    for (int e = 0; e < 8; ++e) {
        int idx = tid + e * 256;              // float2 index, 32 per row
        int off = (idx >> 5) * DP + (idx & 31) * 2;
        v2f v = *(const v2f*)(ybuf + off);
        #pragma unroll
        for (int j = 0; j < N; ++j) {
            v2f k = *(const v2f*)(ks[j] + off);
            v.x = fmaf(cs[j], k.x, v.x);
            v.y = fmaf(cs[j], k.y, v.y);
        }
        *(v2f*)(dst + off) = v;
    }
}

// Fused stage-6 prep: one read of y,k1..k5 produces both
//   ytmp  = y + C61 k1 + C62 k2 + C63 k3 + C64 k4 + C65 k5
//   k1buf = y + DB1 k1 + DB3 k3 + DB4 k4 + DB5 k5   (y-partial, overwrites k1)
__device__ __forceinline__ void stage6_prep(
    float* __restrict__ ytmp, float* __restrict__ k1buf,
    const float* __restrict__ ybuf,
    const float* __restrict__ k2b, const float* __restrict__ k3b,
    const float* __restrict__ k4b, const float* __restrict__ k5b, int tid)
{
    #pragma unroll 2
    for (int e = 0; e < 8; ++e) {
        int idx = tid + e * 256;
        int off = (idx >> 5) * DP + (idx & 31) * 2;
        v2f yv = *(const v2f*)(ybuf + off);
        v2f k1 = *(const v2f*)(k1buf + off);
        v2f k2 = *(const v2f*)(k2b + off);
        v2f k3 = *(const v2f*)(k3b + off);
        v2f k4 = *(const v2f*)(k4b + off);
        v2f k5 = *(const v2f*)(k5b + off);
        v2f t, p;
        t.x = yv.x; t.y = yv.y;
        t.x = fmaf(C61, k1.x, t.x); t.y = fmaf(C61, k1.y, t.y);
        t.x = fmaf(C62, k2.x, t.x); t.y = fmaf(C62, k2.y, t.y);
        t.x = fmaf(C63, k3.x, t.x); t.y = fmaf(C63, k3.y, t.y);
        t.x = fmaf(C64, k4.x, t.x); t.y = fmaf(C64, k4.y, t.y);
        t.x = fmaf(C65, k5.x, t.x); t.y = fmaf(C65, k5.y, t.y);
        p.x = yv.x; p.y = yv.y;
        p.x = fmaf(DB1, k1.x, p.x); p.y = fmaf(DB1, k1.y, p.y);
        p.x = fmaf(DB3, k3.x, p.x); p.y = fmaf(DB3, k3.y, p.y);
        p.x = fmaf(DB4, k4.x, p.x); p.y = fmaf(DB4, k4.y, p.y);
        p.x = fmaf(DB5, k5.x, p.x); p.y = fmaf(DB5, k5.y, p.y);
        *(v2f*)(ytmp + off) = t;
        *(v2f*)(k1buf + off) = p;   // same-thread read-then-write of same element
    }
}

__global__ __launch_bounds__(256, 2) void ode_dopri5_kernel(
    const float* __restrict__ init_pos,  // [16384,64]
    const float* __restrict__ W1,        // [64,128]
    const float* __restrict__ b1,        // [128]
    const float* __restrict__ W2,        // [128,64]
    const float* __restrict__ b2,        // [64]
    float*       __restrict__ out)       // [5,16384,64]
{
    __shared__ float ybuf[BM * DP];
    __shared__ float ytmp[BM * DP];
    __shared__ float hbuf[BM * HP];
    __shared__ float kbuf[5][BM * DP];   // k1..k5 (k1 slot reused for y-partial)

    const int tid = threadIdx.x;
    const int w    = tid >> 5;
    const int lane = tid & 31;
    const int lo = lane & 15;
    const int hi = lane >> 4;
    const int n0  = w * 16;          // GEMM1 N-tile (8 waves cover H=128)
    const int n0b = (w & 3) * 16;    // GEMM2 N-tile (4 n-tiles cover D=64)
    const int mtb = (w >> 2) * 2;    // GEMM2 M-tile pair

    // --- load weight B-fragments into registers (persistent for whole kernel) ---
    v2f w1f[16];
    v2f w2f[32];
    #pragma unroll
    for (int kb = 0; kb < 16; ++kb) {
        int k4 = kb * 4 + 2 * hi;
        w1f[kb].x = W1[k4 * HH + n0 + lo];
        w1f[kb].y = W1[(k4 + 1) * HH + n0 + lo];
    }
    #pragma unroll
    for (int kb = 0; kb < 32; ++kb) {
        int k4 = kb * 4 + 2 * hi;
        w2f[kb].x = W2[k4 * DD + n0b + lo];
        w2f[kb].y = W2[(k4 + 1) * DD + n0b + lo];
    }
    const float bias1 = b1[n0 + lo];
    const float bias2 = b2[n0b + lo];

    // --- load initial state tile into LDS ---
    const long bm0 = (long)blockIdx.x * BM;
    #pragma unroll 2
    for (int e = 0; e < 8; ++e) {
        int f2 = tid + e * 256;          // float2 index: 32 per row
        int row = f2 >> 5;
        int c2  = f2 & 31;
        *(v2f*)&ybuf[row * DP + c2 * 2] =
            *(const v2f*)&init_pos[(bm0 + row) * DD + c2 * 2];
    }

    #pragma unroll 1
    for (int seg = 0; seg < NSEG; ++seg) {
        #pragma unroll 1
        for (int st = 0; st < STEPS_PER_SEG; ++st) {
            // k1 = f(y)
            feval<false>(ybuf, kbuf[0], hbuf, w1f, w2f, bias1, bias2,
                         lo, hi, n0, n0b, mtb, nullptr, 0.0f);
            { const float* ks[1] = {kbuf[0]};
              const float  cs[1] = {C21};
              combine<1>(ytmp, ybuf, ks, cs, tid); }
            // k2
            feval<false>(ytmp, kbuf[1], hbuf, w1f, w2f, bias1, bias2,
                         lo, hi, n0, n0b, mtb, nullptr, 0.0f);
            { const float* ks[2] = {kbuf[0], kbuf[1]};
              const float  cs[2] = {C31, C32};
              combine<2>(ytmp, ybuf, ks, cs, tid); }
            // k3
            feval<false>(ytmp, kbuf[2], hbuf, w1f, w2f, bias1, bias2,
                         lo, hi, n0, n0b, mtb, nullptr, 0.0f);
            { const float* ks[3] = {kbuf[0], kbuf[1], kbuf[2]};
              const float  cs[3] = {C41, C42, C43};
              combine<3>(ytmp, ybuf, ks, cs, tid); }
            // k4
            feval<false>(ytmp, kbuf[3], hbuf, w1f, w2f, bias1, bias2,
                         lo, hi, n0, n0b, mtb, nullptr, 0.0f);
            { const float* ks[4] = {kbuf[0], kbuf[1], kbuf[2], kbuf[3]};
              const float  cs[4] = {C51, C52, C53, C54};
              combine<4>(ytmp, ybuf, ks, cs, tid); }
            // k5
            feval<false>(ytmp, kbuf[4], hbuf, w1f, w2f, bias1, bias2,
                         lo, hi, n0, n0b, mtb, nullptr, 0.0f);
            // fused: ytmp for stage 6 AND y-partial (into k1 slot)
            stage6_prep(ytmp, kbuf[0], ybuf, kbuf[1], kbuf[2], kbuf[3], kbuf[4], tid);
            // k6 fused into the update: y = y_partial + dt*B6 * f(ytmp)
            feval<true>(ytmp, ybuf, hbuf, w1f, w2f, bias1, bias2,
                        lo, hi, n0, n0b, mtb, kbuf[0], DB6);
        }
        // emit y at this eval point
        #pragma unroll 2
        for (int e = 0; e < 8; ++e) {
            int f2 = tid + e * 256;
            int row = f2 >> 5;
            int c2  = f2 & 31;
            *(v2f*)&out[(long)seg * (16384L * DD) + (bm0 + row) * DD + c2 * 2] =
                *(const v2f*)&ybuf[row * DP + c2 * 2];
        }
        // next feval begins with __syncthreads(), which orders these LDS reads
        // before any later overwrite of ybuf.
    }
}

extern "C" void kernel_launch(void* const* d_in, const int* in_sizes, int n_in,
                              void* d_out, int out_size, void* d_ws, size_t ws_size,
                              hipStream_t stream) {
    const float* init_pos = (const float*)d_in[0];
    const float* W1 = (const float*)d_in[1];
    const float* b1 = (const float*)d_in[2];
    const float* W2 = (const float*)d_in[3];
    const float* b2 = (const float*)d_in[4];
    float* out = (float*)d_out;

    dim3 grid(16384 / BM);   // 256 workgroups
    dim3 block(256);         // 8 wave32s
    hipLaunchKernelGGL(ode_dopri5_kernel, grid, block, 0, stream,
                       init_pos, W1, b1, W2, b2, out);
}